// DiscreteWaveletTransform2D_88734024335379
// MI455X (gfx1250) — compile-verified
//
#include <hip/hip_runtime.h>
#include <hip/hip_bf16.h>
#include <stdint.h>

// 2D Haar DWT, single level, float32.
// Input:  x  (8,64,512,512)  -> 512 planes of 512x512
// Output: ll,lh,hl,hh each (8,64,256,256), concatenated flat in d_out.
//
// Memory-bound (AI ~ 0.37 flop/byte, ~1 GiB traffic -> ~46 us floor at
// 23.3 TB/s). Optimize for coalesced b128 traffic, non-temporal hints, and
// the CDNA5 async global->LDS data mover (ASYNCcnt path).

typedef float f32x4 __attribute__((ext_vector_type(4)));
// Match the builtin's parameter type exactly: vector_size(16) int.
typedef int   i32x4v __attribute__((vector_size(16)));
typedef __attribute__((address_space(1))) i32x4v gbl_i32x4;
typedef __attribute__((address_space(3))) i32x4v lds_i32x4;

#if defined(__gfx1250__) &&                                             \
    __has_builtin(__builtin_amdgcn_global_load_async_to_lds_b128) &&    \
    __has_builtin(__builtin_amdgcn_s_wait_asynccnt)
#define DWT_ASYNC_LDS 1
#else
#define DWT_ASYNC_LDS 0
#endif

__global__ __launch_bounds__(256) void
dwt_haar2d_kernel(const float* __restrict__ x, float* __restrict__ out,
                  int planes, size_t quad /* = planes*256*256 */)
{
    const int lane = threadIdx.x;
    const int t  = blockIdx.x * 256 + lane;
    const int tx = t & 63;          // 64 thread-cols, 8 input (4 output) px each
    const int oy = (t >> 6) & 255;  // output row
    const int p  = t >> 14;         // plane (n*64 + c)
    if (p >= planes) return;

    // Input addresses: row pair (2*oy, 2*oy+1), 8 consecutive floats from col tx*8.
    const size_t ibase = ((size_t)p * 512 + (size_t)(oy << 1)) * 512 + (size_t)(tx << 3);
    const float* r0 = x + ibase;        // even row: a/b interleaved
    const float* r1 = r0 + 512;         // odd row : c/d interleaved

    f32x4 r0a, r0b, r1a, r1b;

#if DWT_ASYNC_LDS
    // Stage the 4x16B per thread through LDS via the async data mover.
    __shared__ f32x4 smem[1024];        // 256 threads * 4 vectors = 16 KB
    const unsigned sidx = (unsigned)lane << 2;
    // generic LDS address: low 32 bits == LDS byte offset (aperture rule)
    unsigned l0 = (unsigned)(uintptr_t)(&smem[sidx + 0]);
    unsigned l1 = (unsigned)(uintptr_t)(&smem[sidx + 1]);
    unsigned l2 = (unsigned)(uintptr_t)(&smem[sidx + 2]);
    unsigned l3 = (unsigned)(uintptr_t)(&smem[sidx + 3]);
    __builtin_amdgcn_global_load_async_to_lds_b128(
        (gbl_i32x4*)(uintptr_t)(r0),     (lds_i32x4*)l0, 0, 0);
    __builtin_amdgcn_global_load_async_to_lds_b128(
        (gbl_i32x4*)(uintptr_t)(r0 + 4), (lds_i32x4*)l1, 0, 0);
    __builtin_amdgcn_global_load_async_to_lds_b128(
        (gbl_i32x4*)(uintptr_t)(r1),     (lds_i32x4*)l2, 0, 0);
    __builtin_amdgcn_global_load_async_to_lds_b128(
        (gbl_i32x4*)(uintptr_t)(r1 + 4), (lds_i32x4*)l3, 0, 0);
    __builtin_amdgcn_s_wait_asynccnt(0);   // ASYNCcnt==0 -> LDS writes visible
    r0a = smem[sidx + 0];
    r0b = smem[sidx + 1];
    r1a = smem[sidx + 2];
    r1b = smem[sidx + 3];
#else
    // Direct streaming path: non-temporal b128 loads (no reuse, 512MiB >> L2).
    r0a = __builtin_nontemporal_load((const f32x4*)(r0));
    r0b = __builtin_nontemporal_load((const f32x4*)(r0 + 4));
    r1a = __builtin_nontemporal_load((const f32x4*)(r1));
    r1b = __builtin_nontemporal_load((const f32x4*)(r1 + 4));
#endif

    // De-interleave: a = even cols of even row, b = odd cols of even row, etc.
    const f32x4 a = {r0a.x, r0a.z, r0b.x, r0b.z};
    const f32x4 b = {r0a.y, r0a.w, r0b.y, r0b.w};
    const f32x4 c = {r1a.x, r1a.z, r1b.x, r1b.z};
    const f32x4 d = {r1a.y, r1a.w, r1b.y, r1b.w};

    const f32x4 ll = (a + b + c + d) * 0.5f;
    const f32x4 lh = (c + d - a - b) * 0.5f;
    const f32x4 hl = (b + d - a - c) * 0.5f;
    const f32x4 hh = (a + d - b - c) * 0.5f;

    // Output: 4 contiguous quadrants, each planes*256*256 floats.
    const size_t obase = ((size_t)p << 16) + ((size_t)oy << 8) + (size_t)(tx << 2);
    float* o = out + obase;
    __builtin_nontemporal_store(ll, (f32x4*)(o));
    __builtin_nontemporal_store(lh, (f32x4*)(o + quad));
    __builtin_nontemporal_store(hl, (f32x4*)(o + 2 * quad));
    __builtin_nontemporal_store(hh, (f32x4*)(o + 3 * quad));
}

extern "C" void kernel_launch(void* const* d_in, const int* in_sizes, int n_in,
                              void* d_out, int out_size, void* d_ws, size_t ws_size,
                              hipStream_t stream) {
    const float* x = (const float*)d_in[0];
    float* out     = (float*)d_out;

    const int planes  = in_sizes[0] / (512 * 512);       // 8*64 = 512
    const size_t quad = (size_t)planes * 256 * 256;      // elements per quadrant

    // threads = planes * 256 output rows * 64 thread-cols
    const long long total = (long long)planes * 256 * 64;
    const int blocks = (int)((total + 255) / 256);       // 32768 for full size

    dwt_haar2d_kernel<<<blocks, 256, 0, stream>>>(x, out, planes, quad);
}